// LUTInterpolator_36653250904410
// MI455X (gfx1250) — compile-verified
//
#include <hip/hip_runtime.h>
#include <stdint.h>

// ---------------------------------------------------------------------------
// Trilinear 3D LUT apply for MI455X (gfx1250).
//
// Memory-bound workload: ~400MB HBM streaming (17us floor @ 23.3 TB/s), LUT
// (3.3MB raw / 4.4MB padded) is L2-resident. No matmul structure -> WMMA is
// inapplicable; the CDNA5 paths that pay off are the async-to-LDS streaming
// pipeline (ASYNCcnt), aligned b128 gathers from a float4-padded LUT, and
// non-temporal stores.
// ---------------------------------------------------------------------------

#define LUT_S   65
#define LUT_ENT (LUT_S * LUT_S * LUT_S)   // 274625 entries
#define BLOCK   256
#define TILES   8                          // tiles per block (double-buffered)

typedef float f4  __attribute__((ext_vector_type(4)));
typedef int   gv4i __attribute__((vector_size(16)));   // matches builtin proto

typedef __attribute__((address_space(1))) gv4i* g_v4i_ptr;  // global int4*
typedef __attribute__((address_space(3))) gv4i* l_v4i_ptr;  // LDS int4*

#if __has_builtin(__builtin_amdgcn_global_load_async_to_lds_b128)
#define HAVE_ASYNC 1
#else
#define HAVE_ASYNC 0
#endif

#if __has_builtin(__builtin_amdgcn_s_wait_asynccnt)
#define WAIT_ASYNC(n) __builtin_amdgcn_s_wait_asynccnt(n)
#else
#define WAIT_ASYNC(n) asm volatile("s_wait_asynccnt %0" ::"i"(n) : "memory")
#endif

__device__ __forceinline__ int clampi(int v, int lo, int hi) {
  return v < lo ? lo : (v > hi ? hi : v);
}

// One pixel: faithful to the reference (frac from unclamped floor, indices
// clamped to [0, s-2], lerp a*(1-w) + b*w in z, then y, then x).
template <bool PADDED>
__device__ __forceinline__ void lut_pixel(const void* __restrict__ lutp,
                                          float r, float g, float b,
                                          float* __restrict__ o) {
  r *= (float)(LUT_S - 1);
  g *= (float)(LUT_S - 1);
  b *= (float)(LUT_S - 1);
  float fr = __builtin_floorf(r), fg = __builtin_floorf(g), fb = __builtin_floorf(b);
  float wx = r - fr, wy = g - fg, wz = b - fb;
  int x = clampi((int)fr, 0, LUT_S - 2);
  int y = clampi((int)fg, 0, LUT_S - 2);
  int z = clampi((int)fb, 0, LUT_S - 2);
  int base = (x * LUT_S + y) * LUT_S + z;
  const int DY = LUT_S;           // +1 in y
  const int DX = LUT_S * LUT_S;   // +1 in x
  float omx = 1.0f - wx, omy = 1.0f - wy, omz = 1.0f - wz;

  if (PADDED) {
    // float4-padded LUT: every tap is an aligned global_load_b128 gather.
    const f4* lut4 = (const f4*)lutp;
    f4 c000 = lut4[base],           c001 = lut4[base + 1];
    f4 c010 = lut4[base + DY],      c011 = lut4[base + DY + 1];
    f4 c100 = lut4[base + DX],      c101 = lut4[base + DX + 1];
    f4 c110 = lut4[base + DX + DY], c111 = lut4[base + DX + DY + 1];
    f4 v00 = c000 * omz + c001 * wz;
    f4 v01 = c010 * omz + c011 * wz;
    f4 v10 = c100 * omz + c101 * wz;
    f4 v11 = c110 * omz + c111 * wz;
    f4 v0 = v00 * omy + v01 * wy;
    f4 v1 = v10 * omy + v11 * wy;
    f4 v  = v0 * omx + v1 * wx;
    o[0] = v.x; o[1] = v.y; o[2] = v.z;
  } else {
    // Raw 3-float entries: z and z+1 are 6 contiguous floats per (x,y) base.
    const float* lut = (const float*)lutp;
    float c00[6], c01[6], c10[6], c11[6];
    __builtin_memcpy(c00, lut + (size_t)base * 3, 24);
    __builtin_memcpy(c01, lut + (size_t)(base + DY) * 3, 24);
    __builtin_memcpy(c10, lut + (size_t)(base + DX) * 3, 24);
    __builtin_memcpy(c11, lut + (size_t)(base + DX + DY) * 3, 24);
#pragma unroll
    for (int ch = 0; ch < 3; ++ch) {
      float v00 = c00[ch] * omz + c00[ch + 3] * wz;
      float v01 = c01[ch] * omz + c01[ch + 3] * wz;
      float v10 = c10[ch] * omz + c10[ch + 3] * wz;
      float v11 = c11[ch] * omz + c11[ch + 3] * wz;
      float v0 = v00 * omy + v01 * wy;
      float v1 = v10 * omy + v11 * wy;
      o[ch] = v0 * omx + v1 * wx;
    }
  }
}

// 4 pixels (12 floats = 3 x b128) per thread, NT stores.
template <bool PADDED>
__device__ __forceinline__ void process_group(const void* __restrict__ lutp,
                                              float* __restrict__ out, int group,
                                              f4 p0, f4 p1, f4 p2) {
  float pix[12] = {p0.x, p0.y, p0.z, p0.w, p1.x, p1.y, p1.z, p1.w,
                   p2.x, p2.y, p2.z, p2.w};
  float res[12];
#pragma unroll
  for (int p = 0; p < 4; ++p)
    lut_pixel<PADDED>(lutp, pix[3 * p], pix[3 * p + 1], pix[3 * p + 2], res + 3 * p);
  f4 r0 = {res[0], res[1], res[2],  res[3]};
  f4 r1 = {res[4], res[5], res[6],  res[7]};
  f4 r2 = {res[8], res[9], res[10], res[11]};
  f4* o = (f4*)out + (size_t)group * 3;
  __builtin_nontemporal_store(r0, o);
  __builtin_nontemporal_store(r1, o + 1);
  __builtin_nontemporal_store(r2, o + 2);
}

template <bool PADDED>
__global__ __launch_bounds__(BLOCK) void lut3d_apply(
    const float* __restrict__ rgb, const void* __restrict__ lutp,
    float* __restrict__ out, int ngroups) {
  const int tid = threadIdx.x;
#if HAVE_ASYNC
  // Double-buffered async streaming: tile t+1 -> LDS (coalesced, ASYNCcnt)
  // while computing tile t read back from LDS in pixel-major order.
  __shared__ f4 smem[2 * BLOCK * 3];  // 24 KB
  const f4* rgb4 = (const f4*)rgb;
  const int lastF4 = ngroups * 3 - 1;

  auto issue = [&](int t, int buf) {
    int tileF4 = (blockIdx.x * TILES + t) * (BLOCK * 3);
#pragma unroll
    for (int k = 0; k < 3; ++k) {
      int gi = tileF4 + k * BLOCK + tid;          // fully coalesced order
      gi = gi > lastF4 ? lastF4 : gi;             // clamp OOB lanes
      __builtin_amdgcn_global_load_async_to_lds_b128(
          (g_v4i_ptr)(rgb4 + gi),
          (l_v4i_ptr)(&smem[buf * (BLOCK * 3) + k * BLOCK + tid]),
          0, 0);
    }
  };

  issue(0, 0);
#pragma unroll 1
  for (int t = 0; t < TILES; ++t) {
    const int buf = t & 1;
    if (t + 1 < TILES) {
      issue(t + 1, buf ^ 1);
      WAIT_ASYNC(3);   // our 3 in flight for t+1; tile t's 3 are complete
    } else {
      WAIT_ASYNC(0);
    }
    __syncthreads();   // all waves' tile-t data resident in LDS
    f4 p0 = smem[buf * (BLOCK * 3) + tid * 3 + 0];
    f4 p1 = smem[buf * (BLOCK * 3) + tid * 3 + 1];
    f4 p2 = smem[buf * (BLOCK * 3) + tid * 3 + 2];
    __syncthreads();   // reads done before tile t+2 overwrites this buffer
    int group = (blockIdx.x * TILES + t) * BLOCK + tid;
    if (group < ngroups) process_group<PADDED>(lutp, out, group, p0, p1, p2);
  }
#else
  // Fallback: direct NT streaming loads (still b128, just no async pipeline).
#pragma unroll 1
  for (int t = 0; t < TILES; ++t) {
    int group = (blockIdx.x * TILES + t) * BLOCK + tid;
    if (group < ngroups) {
      const f4* ip = (const f4*)rgb + (size_t)group * 3;
      f4 p0 = __builtin_nontemporal_load(ip);
      f4 p1 = __builtin_nontemporal_load(ip + 1);
      f4 p2 = __builtin_nontemporal_load(ip + 2);
      process_group<PADDED>(lutp, out, group, p0, p1, p2);
    }
  }
#endif
}

// Scalar tail for pixel counts not divisible by 4 (robustness).
template <bool PADDED>
__global__ void lut3d_tail(const float* __restrict__ rgb,
                           const void* __restrict__ lutp,
                           float* __restrict__ out, int startPix, int npix) {
  int i = startPix + blockIdx.x * blockDim.x + threadIdx.x;
  if (i < npix) {
    float o[3];
    lut_pixel<PADDED>(lutp, rgb[3 * i], rgb[3 * i + 1], rgb[3 * i + 2], o);
    out[3 * i] = o[0];
    out[3 * i + 1] = o[1];
    out[3 * i + 2] = o[2];
  }
}

// Repack 3-float LUT entries into 16B-aligned float4 entries in d_ws.
__global__ void lut_repack(const float* __restrict__ lut, f4* __restrict__ lut4) {
  int i = blockIdx.x * blockDim.x + threadIdx.x;
  if (i < LUT_ENT) {
    f4 v = {lut[3 * i], lut[3 * i + 1], lut[3 * i + 2], 0.0f};
    lut4[i] = v;
  }
}

extern "C" void kernel_launch(void* const* d_in, const int* in_sizes, int n_in,
                              void* d_out, int out_size, void* d_ws, size_t ws_size,
                              hipStream_t stream) {
  const float* rgb = (const float*)d_in[0];
  const float* lut = (const float*)d_in[1];
  float* out = (float*)d_out;

  int npix = out_size / 3;
  int ngroups = npix / 4;
  int tailStart = ngroups * 4;
  int blocks = (ngroups + BLOCK * TILES - 1) / (BLOCK * TILES);

  bool padded = ws_size >= (size_t)LUT_ENT * sizeof(f4);
  if (padded) {
    f4* lut4 = (f4*)d_ws;
    lut_repack<<<(LUT_ENT + 255) / 256, 256, 0, stream>>>(lut, lut4);
    if (blocks > 0)
      lut3d_apply<true><<<blocks, BLOCK, 0, stream>>>(rgb, lut4, out, ngroups);
    if (tailStart < npix)
      lut3d_tail<true><<<1, 64, 0, stream>>>(rgb, lut4, out, tailStart, npix);
  } else {
    if (blocks > 0)
      lut3d_apply<false><<<blocks, BLOCK, 0, stream>>>(rgb, lut, out, ngroups);
    if (tailStart < npix)
      lut3d_tail<false><<<1, 64, 0, stream>>>(rgb, lut, out, tailStart, npix);
  }
}